// DeepInstructedAttentionPositionScores_72189810311268
// MI455X (gfx1250) — compile-verified
//
#include <hip/hip_runtime.h>

typedef __attribute__((ext_vector_type(2))) float v2f;
typedef __attribute__((ext_vector_type(4))) float vf4;
typedef __attribute__((ext_vector_type(8))) float v8f;

constexpr int EMB   = 64;
constexpr int HEADS = 8;
constexpr int CAP   = 32;
constexpr int TBL   = 2 * CAP - 1;   // 63
constexpr int DIMQ  = 4116;
constexpr int DIMI  = 20;
// dim_h = dim_w = dim_d = 16, S = 4096, DIMQ = DIMI + S

// workspace layout (float offsets)
constexpr int WS_CROSS = 0;              // 8*20
constexpr int WS_ROW   = 160;            // 8*256
constexpr int WS_COL   = 160 + 2048;     // 2208
constexpr int WS_DEP   = 2208 + 2048;    // 4256
constexpr int WS_TOT   = 4256 + 2048;    // 6304 floats (~25 KB)

// ---------------------------------------------------------------------------
// Kernel 1: compute cross_s [8,20] and row/col/dep_s [8,256] with f32 WMMA.
// One wave (32 threads) per 16x16 output tile. 50 tiles:
//   blocks 0..15  -> row_s  (enc_h,  w_h)
//   blocks 16..31 -> col_s  (enc_w,  w_w)
//   blocks 32..47 -> dep_s  (enc_d,  w_d)
//   blocks 48..49 -> cross_s (enc_cross, w_cross), cols 0..19 valid
// D[h, p] = sum_c W[h,c] * B[c,p], B[c,p] = table[srcrow(p)][c].
// ---------------------------------------------------------------------------
__global__ __launch_bounds__(32)
void dia_setup(const float* __restrict__ enc_cross,
               const float* __restrict__ enc_h,
               const float* __restrict__ enc_w,
               const float* __restrict__ enc_d,
               const float* __restrict__ w_cross,
               const float* __restrict__ w_h,
               const float* __restrict__ w_w,
               const float* __restrict__ w_d,
               float* __restrict__ ws)
{
    const int lane = threadIdx.x;
    const int b    = blockIdx.x;

    const float* W;
    const float* E;
    int tile, outOff, outStride;
    bool isCross;
    if (b < 48) {
        const int t = b >> 4;
        tile = b & 15; isCross = false; outStride = 256;
        if (t == 0)      { W = w_h; E = enc_h; outOff = WS_ROW; }
        else if (t == 1) { W = w_w; E = enc_w; outOff = WS_COL; }
        else             { W = w_d; E = enc_d; outOff = WS_DEP; }
    } else {
        tile = b - 48; isCross = true; outStride = DIMI;
        W = w_cross; E = enc_cross; outOff = WS_CROSS;
    }

    // A-matrix 16x4 f32 layout: lanes 0-15 -> M=lane, K={0,1}; lanes 16-31 -> M=lane-16, K={2,3}
    const int M    = lane & 15;
    const int koff = (lane < 16) ? 0 : 2;
    const float amask = (M < HEADS) ? 1.0f : 0.0f;
    const int   Mc    = M & 7;

    // B-matrix 4x16: column index per lane
    const int N = lane & 15;
    const int p = tile * 16 + N;
    int   brow;
    float bmask = 1.0f;
    if (!isCross) {
        const int n = p >> 4, m = p & 15;
        int idx = m - n + (CAP - 1);
        idx = idx < 0 ? 0 : (idx > TBL - 1 ? TBL - 1 : idx);
        brow = idx;                              // rel-embed table row
    } else {
        brow  = (p < DIMI) ? p : 0;              // cross_emb row (flat [20,64])
        bmask = (p < DIMI) ? 1.0f : 0.0f;
    }

    v8f acc = {};
    #pragma unroll
    for (int kk = 0; kk < 16; ++kk) {
        const int c0 = kk * 4 + koff;
        v2f a, bb;
        a.x  = W[Mc * EMB + c0]     * amask;
        a.y  = W[Mc * EMB + c0 + 1] * amask;
        bb.x = E[brow * EMB + c0]     * bmask;
        bb.y = E[brow * EMB + c0 + 1] * bmask;
        acc = __builtin_amdgcn_wmma_f32_16x16x4_f32(
            /*neg_a=*/false, a, /*neg_b=*/false, bb,
            /*c_mod=*/(short)0, acc, /*reuse_a=*/false, /*reuse_b=*/false);
    }

    // C/D layout: VGPR r, lanes 0-15 hold M=r (the 8 valid heads); lanes 16-31 hold M=r+8 (padding)
    if (lane < 16) {
        const int col = tile * 16 + lane;
        if (!isCross || col < DIMI) {
            #pragma unroll
            for (int r = 0; r < 8; ++r)
                ws[outOff + r * outStride + col] = acc[r];
        }
    }
}

// ---------------------------------------------------------------------------
// Kernel 2: stream out [8, 4116, 4116] f32 (542 MB). Pure store-BW bound.
// blockIdx.y = head. Tables live in LDS; non-temporal 128-bit stores.
// ---------------------------------------------------------------------------
__global__ __launch_bounds__(256)
void dia_fill(const float* __restrict__ ws, float* __restrict__ out)
{
    __shared__ float lds[WS_TOT];
    for (int i = threadIdx.x; i < WS_TOT; i += 256) lds[i] = ws[i];
    __syncthreads();

    const int h = blockIdx.y;
    const float* __restrict__ crossT = &lds[WS_CROSS + h * DIMI];
    const float* __restrict__ rowT   = &lds[WS_ROW   + h * 256];
    const float* __restrict__ colT   = &lds[WS_COL   + h * 256];
    const float* __restrict__ depT   = &lds[WS_DEP   + h * 256];

    constexpr int ROWF4 = DIMQ / 4;              // 1029 float4 per row
    constexpr int PER_H = DIMQ * ROWF4;          // 4,235,364 float4 per head
    const float invT = 0.125f;                   // 64^-0.5
    const float c3   = 0.125f / 3.0f;

    vf4* __restrict__ outv = reinterpret_cast<vf4*>(out) + (size_t)h * PER_H;

    for (int t = blockIdx.x * 256 + threadIdx.x; t < PER_H; t += gridDim.x * 256) {
        const int q = t / ROWF4;
        const int f = t - q * ROWF4;
        vf4 val;
        if (q < DIMI) {
            val = (vf4){0.f, 0.f, 0.f, 0.f};                 // rows < dim_i stay zero
        } else if (f < 5) {
            const int k0 = f * 4;                            // cross block, cols 0..19
            val.x = invT * crossT[k0];
            val.y = invT * crossT[k0 + 1];
            val.z = invT * crossT[k0 + 2];
            val.w = invT * crossT[k0 + 3];
        } else {
            const int r  = q - DIMI;                         // content row (i,j,k)
            const int c0 = f * 4 - DIMI;                     // content col (l,m,n), c0 % 4 == 0
            const float base = rowT[((r >> 8) << 4) + (c0 >> 8)]
                             + colT[((r >> 4) & 15) * 16 + ((c0 >> 4) & 15)];
            const vf4 dv = *reinterpret_cast<const vf4*>(&depT[((r & 15) << 4) + (c0 & 15)]);
            val.x = c3 * (base + dv.x);
            val.y = c3 * (base + dv.y);
            val.z = c3 * (base + dv.z);
            val.w = c3 * (base + dv.w);
        }
        __builtin_nontemporal_store(val, &outv[t]);
    }
}

extern "C" void kernel_launch(void* const* d_in, const int* in_sizes, int n_in,
                              void* d_out, int out_size, void* d_ws, size_t ws_size,
                              hipStream_t stream) {
    const float* enc_cross = (const float*)d_in[0];
    const float* enc_h     = (const float*)d_in[1];
    const float* enc_w     = (const float*)d_in[2];
    const float* enc_d     = (const float*)d_in[3];
    const float* w_cross   = (const float*)d_in[4];
    const float* w_h       = (const float*)d_in[5];
    const float* w_w       = (const float*)d_in[6];
    const float* w_d       = (const float*)d_in[7];
    float* ws  = (float*)d_ws;
    float* out = (float*)d_out;

    dia_setup<<<50, 32, 0, stream>>>(enc_cross, enc_h, enc_w, enc_d,
                                     w_cross, w_h, w_w, w_d, ws);

    dim3 grid(2048, HEADS);
    dia_fill<<<grid, 256, 0, stream>>>(ws, out);
}